// MambaEncoder_6262062318301
// MI455X (gfx1250) — compile-verified
//
#include <hip/hip_runtime.h>
#include <hip/hip_bf16.h>

#define D_   512
#define DI_  1024
#define S_   16
#define R_   32
#define K_   4
#define NL_  6
#define B_   4
#define L_   1024

typedef float v2f __attribute__((ext_vector_type(2)));
typedef float v8f __attribute__((ext_vector_type(8)));
typedef unsigned u32x4 __attribute__((ext_vector_type(4)));
typedef unsigned u32x8 __attribute__((ext_vector_type(8)));

__device__ __forceinline__ float softplusf(float v) {
    return v > 20.f ? v : log1pf(expf(v));
}
__device__ __forceinline__ float siluf(float v) {
    return v / (1.f + expf(-v));
}

// ---------------- block reduction helper (wave32) ----------------
__device__ __forceinline__ float block_sum(float v, float* smem) {
    #pragma unroll
    for (int o = 16; o > 0; o >>= 1) v += __shfl_xor(v, o);
    int lane = threadIdx.x & 31;
    int wid  = threadIdx.x >> 5;
    if (lane == 0) smem[wid] = v;
    __syncthreads();
    float s = 0.f;
    int nw = blockDim.x >> 5;
    for (int i = 0; i < nw; ++i) s += smem[i];
    __syncthreads();
    return s;
}

// ---------------- TDM: async 2D tile load global->LDS ----------------
// Loads a (rows x k_elems) f32 tile whose rows are stride_elems apart,
// starting at gptr, into LDS at byte offset lds_off (packed rows*k_elems).
// D# layout per CDNA5 ISA ch.8 (group0 128b, group1 256b), data_size=4B.
__device__ __forceinline__ void tdm_load_2d(unsigned lds_off, const float* gptr,
                                            unsigned rows, unsigned k_elems,
                                            unsigned stride_elems) {
    unsigned long long ga = (unsigned long long)(uintptr_t)gptr;
    u32x4 g0;
    g0[0] = 1u;                                      // count=1, user mode
    g0[1] = lds_off;                                 // lds_addr
    g0[2] = (unsigned)(ga & 0xffffffffu);            // global_addr[31:0]
    g0[3] = (unsigned)(ga >> 32) | (2u << 30);       // global_addr[56:32] | type=2
    u32x8 g1;
    g1[0] = 2u << 16;                                // workgroup_mask=0, data_size=2 (4B)
    g1[1] = (stride_elems & 0xffffu) << 16;          // tensor_dim0[15:0] (>= tile)
    g1[2] = (stride_elems >> 16) | (rows << 16);     // tensor_dim0[31:16] | tensor_dim1[15:0]
    g1[3] = k_elems << 16;                           // tensor_dim1[31:16]=0 | tile_dim0
    g1[4] = rows;                                    // tile_dim1 (tile_dim2=0)
    g1[5] = stride_elems;                            // tensor_dim0_stride[31:0]
    g1[6] = 0;                                       // stride0[47:32]=0, stride1[15:0]=0
    g1[7] = 0;
    asm volatile("tensor_load_to_lds %0, %1" :: "s"(g0), "s"(g1) : "memory");
}

// ---------------- residual add + LayerNorm ----------------
// one block (128 thr) per row of D_=512
__global__ void add_ln_kernel(const float* hid, float* resid,
                              const float* __restrict__ w, const float* __restrict__ b,
                              float* out, int first) {
    __shared__ float smem[8];
    int row = blockIdx.x;
    size_t base = (size_t)row * D_;
    int c = threadIdx.x * 4;
    float4 hv = *(const float4*)(hid + base + c);
    float4 rv;
    if (first) rv = hv;
    else {
        rv = *(const float4*)(resid + base + c);
        rv.x += hv.x; rv.y += hv.y; rv.z += hv.z; rv.w += hv.w;
    }
    *(float4*)(resid + base + c) = rv;
    float mean = block_sum(rv.x + rv.y + rv.z + rv.w, smem) * (1.f / D_);
    float dx = rv.x - mean, dy = rv.y - mean, dz = rv.z - mean, dw = rv.w - mean;
    float var = block_sum(dx*dx + dy*dy + dz*dz + dw*dw, smem) * (1.f / D_);
    float rstd = rsqrtf(var + 1e-5f);
    float4 wv = *(const float4*)(w + c);
    float4 bv = *(const float4*)(b + c);
    float4 ov;
    ov.x = dx * rstd * wv.x + bv.x;
    ov.y = dy * rstd * wv.y + bv.y;
    ov.z = dz * rstd * wv.z + bv.z;
    ov.w = dw * rstd * wv.w + bv.w;
    *(float4*)(out + base + c) = ov;
}

// ---------------- fp32 WMMA GEMM with TDM double-buffered LDS staging ----------
// C[M,N] = A[M,Kd] * W[N,Kd]^T (+bias, softplus).  block = 128 thr (4 waves),
// each wave a 32x32 tile, block tiles 64x64, k-tiles of 32.  grid = (N/64, M/64)
template <int EPI>
__global__ void gemm_wmma(const float* __restrict__ A, int lda,
                          const float* __restrict__ W, int Kd,
                          const float* __restrict__ bias,
                          float* __restrict__ C, int N) {
    __shared__ float sA[2][64 * 32];
    __shared__ float sB[2][64 * 32];
    int lane = threadIdx.x & 31;
    int wid  = threadIdx.x >> 5;
    int r    = lane & 15;
    int koff = (lane >> 4) << 1;               // 0 or 2
    int blockM = blockIdx.y * 64;
    int blockN = blockIdx.x * 64;
    int wm = (wid >> 1) * 32;
    int wn = (wid & 1) * 32;

    unsigned sAoff[2], sBoff[2];
    sAoff[0] = (unsigned)(uintptr_t)&sA[0][0];
    sAoff[1] = (unsigned)(uintptr_t)&sA[1][0];
    sBoff[0] = (unsigned)(uintptr_t)&sB[0][0];
    sBoff[1] = (unsigned)(uintptr_t)&sB[1][0];

    const bool issuer = (wid == 0);
    int nkt = Kd >> 5;                         // k-tiles of 32

    if (issuer) {
        tdm_load_2d(sAoff[0], A + (size_t)blockM * lda, 64, 32, (unsigned)lda);
        tdm_load_2d(sBoff[0], W + (size_t)blockN * Kd,  64, 32, (unsigned)Kd);
    }

    v8f acc00 = {}; v8f acc01 = {}; v8f acc10 = {}; v8f acc11 = {};

    for (int kt = 0; kt < nkt; ++kt) {
        int cur = kt & 1;
        if (issuer) {
            if (kt + 1 < nkt) {
                int nxt = cur ^ 1;
                tdm_load_2d(sAoff[nxt], A + (size_t)blockM * lda + (kt + 1) * 32,
                            64, 32, (unsigned)lda);
                tdm_load_2d(sBoff[nxt], W + (size_t)blockN * Kd + (kt + 1) * 32,
                            64, 32, (unsigned)Kd);
                __builtin_amdgcn_s_wait_tensorcnt(2);  // current tile's 2 loads done
            } else {
                __builtin_amdgcn_s_wait_tensorcnt(0);
            }
        }
        __syncthreads();                        // tile kt visible to all waves

        const float* pA0 = &sA[cur][(wm + r)      * 32];
        const float* pA1 = &sA[cur][(wm + 16 + r) * 32];
        const float* pB0 = &sB[cur][(wn + r)      * 32];
        const float* pB1 = &sB[cur][(wn + 16 + r) * 32];
        #pragma unroll
        for (int k = 0; k < 32; k += 4) {
            v2f a0 = *(const v2f*)(pA0 + k + koff);
            v2f a1 = *(const v2f*)(pA1 + k + koff);
            v2f b0 = *(const v2f*)(pB0 + k + koff);
            v2f b1 = *(const v2f*)(pB1 + k + koff);
            acc00 = __builtin_amdgcn_wmma_f32_16x16x4_f32(false, a0, false, b0, (short)0, acc00, false, false);
            acc01 = __builtin_amdgcn_wmma_f32_16x16x4_f32(false, a0, false, b1, (short)0, acc01, false, false);
            acc10 = __builtin_amdgcn_wmma_f32_16x16x4_f32(false, a1, false, b0, (short)0, acc10, false, false);
            acc11 = __builtin_amdgcn_wmma_f32_16x16x4_f32(false, a1, false, b1, (short)0, acc11, false, false);
        }
        __syncthreads();                        // reads done before buffer reuse
    }

    int tileM = blockM + wm;
    int tileN = blockN + wn;
    int col0 = tileN + r;
    int col1 = tileN + 16 + r;
    float bias0 = 0.f, bias1 = 0.f;
    if (EPI == 1) { bias0 = bias[col0]; bias1 = bias[col1]; }
    int rbase = tileM + ((lane >> 4) << 3);

    #pragma unroll
    for (int i = 0; i < 8; ++i) {
        int row0 = rbase + i;
        int row1 = row0 + 16;
        float v00 = acc00[i], v01 = acc01[i], v10 = acc10[i], v11 = acc11[i];
        if (EPI == 1) {
            v00 = softplusf(v00 + bias0);
            v01 = softplusf(v01 + bias1);
            v10 = softplusf(v10 + bias0);
            v11 = softplusf(v11 + bias1);
        }
        C[(size_t)row0 * N + col0] = v00;
        C[(size_t)row0 * N + col1] = v01;
        C[(size_t)row1 * N + col0] = v10;
        C[(size_t)row1 * N + col1] = v11;
    }
}

// ---------------- causal conv (K=4) + SiLU on x-half of xz ----------------
__global__ void conv_silu_kernel(const float* __restrict__ xz,
                                 const float* __restrict__ cw,   // (DI,K)
                                 const float* __restrict__ cb,   // (DI)
                                 float* __restrict__ xbuf) {
    size_t idx = (size_t)blockIdx.x * blockDim.x + threadIdx.x;  // B*L*DI
    int c = (int)(idx & (DI_ - 1));
    size_t bl = idx >> 10;                                       // b*L + l
    int l = (int)(bl & (L_ - 1));
    size_t base = bl - l;                                        // b*L
    const float w0 = cw[c*4+0], w1 = cw[c*4+1], w2 = cw[c*4+2], w3 = cw[c*4+3];
    float acc = cb[c] + w3 * xz[(base + l) * (size_t)(2*DI_) + c];
    if (l > 0) acc += w2 * xz[(base + l - 1) * (size_t)(2*DI_) + c];
    if (l > 1) acc += w1 * xz[(base + l - 2) * (size_t)(2*DI_) + c];
    if (l > 2) acc += w0 * xz[(base + l - 3) * (size_t)(2*DI_) + c];
    xbuf[idx] = siluf(acc);
}

// ---------------- selective scan: 16 lanes (states) per (b,d) channel ------
// thread g -> s = g&15, channel cd = g>>4 (b = cd/DI, d = cd%DI).
// y_t reduction over s via 4 shfl_xor hops inside the 16-lane group.
__global__ void scan_kernel(float* __restrict__ dt_y,          // in dt, out y (B*L, DI)
                            const float* __restrict__ xbuf,    // (B*L, DI)
                            const float* __restrict__ dbc,     // (B*L, 64)
                            const float* __restrict__ xz,      // (B*L, 2*DI), z upper half
                            const float* __restrict__ A_log,   // (DI, S)
                            const float* __restrict__ Dp) {    // (DI)
    int g = blockIdx.x * blockDim.x + threadIdx.x;             // B*DI*S = 65536
    int s = g & (S_ - 1);
    int cd = g >> 4;
    int b = cd >> 10;
    int d = cd & (DI_ - 1);
    float An = -expf(A_log[d * S_ + s]);
    float h = 0.f;
    float Dd = Dp[d];
    size_t row0 = (size_t)b * L_;
    for (int t = 0; t < L_; ++t) {
        size_t row = row0 + t;
        float dt_t = dt_y[row * DI_ + d];
        float x_t  = xbuf[row * DI_ + d];
        float Bv   = dbc[row * 64 + R_ + s];
        float Cv   = dbc[row * 64 + R_ + S_ + s];
        h = expf(dt_t * An) * h + dt_t * x_t * Bv;
        float p = h * Cv;
        p += __shfl_xor(p, 1);
        p += __shfl_xor(p, 2);
        p += __shfl_xor(p, 4);
        p += __shfl_xor(p, 8);
        if (s == 0) {
            float z = xz[row * (size_t)(2*DI_) + DI_ + d];
            dt_y[row * DI_ + d] = (p + x_t * Dd) * siluf(z);
        }
    }
}

// ---------------- final: residual add + LN(normf) + LN(after) ----------------
__global__ void final_kernel(const float* __restrict__ hid, const float* __restrict__ resid,
                             const float* __restrict__ w1, const float* __restrict__ b1,
                             const float* __restrict__ w2, const float* __restrict__ b2,
                             float* __restrict__ out) {
    __shared__ float smem[8];
    int row = blockIdx.x;
    size_t base = (size_t)row * D_;
    int c = threadIdx.x * 4;
    float4 hv = *(const float4*)(hid + base + c);
    float4 rv = *(const float4*)(resid + base + c);
    rv.x += hv.x; rv.y += hv.y; rv.z += hv.z; rv.w += hv.w;

    float mean = block_sum(rv.x + rv.y + rv.z + rv.w, smem) * (1.f / D_);
    float dx = rv.x - mean, dy = rv.y - mean, dz = rv.z - mean, dw = rv.w - mean;
    float var = block_sum(dx*dx + dy*dy + dz*dz + dw*dw, smem) * (1.f / D_);
    float rstd = rsqrtf(var + 1e-5f);
    float4 wv = *(const float4*)(w1 + c);
    float4 bv = *(const float4*)(b1 + c);
    float x1 = dx * rstd * wv.x + bv.x;
    float y1 = dy * rstd * wv.y + bv.y;
    float z1 = dz * rstd * wv.z + bv.z;
    float u1 = dw * rstd * wv.w + bv.w;

    float mean2 = block_sum(x1 + y1 + z1 + u1, smem) * (1.f / D_);
    float ex = x1 - mean2, ey = y1 - mean2, ez = z1 - mean2, ew = u1 - mean2;
    float var2 = block_sum(ex*ex + ey*ey + ez*ez + ew*ew, smem) * (1.f / D_);
    float rstd2 = rsqrtf(var2 + 1e-5f);
    float4 wv2 = *(const float4*)(w2 + c);
    float4 bv2 = *(const float4*)(b2 + c);
    float4 ov;
    ov.x = ex * rstd2 * wv2.x + bv2.x;
    ov.y = ey * rstd2 * wv2.y + bv2.y;
    ov.z = ez * rstd2 * wv2.z + bv2.z;
    ov.w = ew * rstd2 * wv2.w + bv2.w;
    *(float4*)(out + base + c) = ov;
}

__global__ void olens_kernel(const int* __restrict__ ilens, int* __restrict__ o) {
    int i = threadIdx.x;
    if (i < B_) o[i] = ilens[i] < L_ ? ilens[i] : L_;
}

extern "C" void kernel_launch(void* const* d_in, const int* in_sizes, int n_in,
                              void* d_out, int out_size, void* d_ws, size_t ws_size,
                              hipStream_t stream) {
    const float* xs_pad    = (const float*)d_in[0];
    const int*   ilens     = (const int*)  d_in[1];
    const float* norm_w    = (const float*)d_in[2];
    const float* norm_b    = (const float*)d_in[3];
    const float* in_proj_w = (const float*)d_in[4];
    const float* conv_w    = (const float*)d_in[5];
    const float* conv_b    = (const float*)d_in[6];
    const float* x_proj_w  = (const float*)d_in[7];
    const float* dt_proj_w = (const float*)d_in[8];
    const float* dt_proj_b = (const float*)d_in[9];
    const float* A_log     = (const float*)d_in[10];
    const float* D_param   = (const float*)d_in[11];
    const float* out_proj_w= (const float*)d_in[12];
    const float* normf_w   = (const float*)d_in[13];
    const float* normf_b   = (const float*)d_in[14];
    const float* after_w   = (const float*)d_in[15];
    const float* after_b   = (const float*)d_in[16];

    const int M = B_ * L_;             // 4096 rows

    float* f = (float*)d_ws;
    size_t o = 0;
    float* resid  = f + o; o += (size_t)M * D_;        // 2M
    float* hn     = f + o; o += (size_t)M * D_;        // 2M
    float* hidden = f + o; o += (size_t)M * D_;        // 2M
    float* xz     = f + o; o += (size_t)M * 2 * DI_;   // 8M
    float* xbuf   = f + o; o += (size_t)M * DI_;       // 4M
    float* dbc    = f + o; o += (size_t)M * 64;        // 256K
    float* dtb    = f + o; o += (size_t)M * DI_;       // 4M

    for (int i = 0; i < NL_; ++i) {
        const float* hid_in = (i == 0) ? xs_pad : hidden;
        add_ln_kernel<<<M, 128, 0, stream>>>(hid_in, resid,
                                             norm_w + (size_t)i * D_, norm_b + (size_t)i * D_,
                                             hn, (i == 0) ? 1 : 0);
        // xz = hn @ in_proj_w^T : (4096, 2048), Kd=512
        gemm_wmma<0><<<dim3(2 * DI_ / 64, M / 64), 128, 0, stream>>>(
            hn, D_, in_proj_w + (size_t)i * 2 * DI_ * D_, D_, nullptr, xz, 2 * DI_);
        // conv + silu -> xbuf
        conv_silu_kernel<<<(M * DI_) / 256, 256, 0, stream>>>(
            xz, conv_w + (size_t)i * DI_ * K_, conv_b + (size_t)i * DI_, xbuf);
        // dbc = xbuf @ x_proj_w^T : (4096, 64), Kd=1024
        gemm_wmma<0><<<dim3(64 / 64, M / 64), 128, 0, stream>>>(
            xbuf, DI_, x_proj_w + (size_t)i * 64 * DI_, DI_, nullptr, dbc, 64);
        // dt = softplus(dbc[:, :R] @ dt_proj_w^T + dt_b) : (4096, 1024), Kd=32
        gemm_wmma<1><<<dim3(DI_ / 64, M / 64), 128, 0, stream>>>(
            dbc, 64, dt_proj_w + (size_t)i * DI_ * R_, R_,
            dt_proj_b + (size_t)i * DI_, dtb, DI_);
        // selective scan: dtb becomes y (in place); 16 lanes per channel
        scan_kernel<<<(B_ * DI_ * S_) / 256, 256, 0, stream>>>(
            dtb, xbuf, dbc, xz, A_log + (size_t)i * DI_ * S_, D_param + (size_t)i * DI_);
        // hidden = y @ out_proj_w^T : (4096, 512), Kd=1024
        gemm_wmma<0><<<dim3(D_ / 64, M / 64), 128, 0, stream>>>(
            dtb, DI_, out_proj_w + (size_t)i * D_ * DI_, DI_, nullptr, hidden, D_);
    }

    float* out = (float*)d_out;
    final_kernel<<<M, 128, 0, stream>>>(hidden, resid, normf_w, normf_b, after_w, after_b, out);
    olens_kernel<<<1, 32, 0, stream>>>(ilens, (int*)(out + (size_t)M * D_));
}